// L2FeaStConv_84859963834436
// MI455X (gfx1250) — compile-verified
//
#include <hip/hip_runtime.h>

#define IN_CH 32
#define HID   400
#define OUT_CH 4

typedef __attribute__((ext_vector_type(2))) float v2f;
typedef __attribute__((ext_vector_type(8))) float v8f;

// ---------------------------------------------------------------------------
// Kernel 1: self-loop init.  s1[n,:] = x[n,:], deg[n] = 1 (self loop).
// ---------------------------------------------------------------------------
__global__ void k_init1(const float* __restrict__ x, float* __restrict__ s1,
                        float* __restrict__ deg, int N) {
    int t = blockIdx.x * blockDim.x + threadIdx.x;
    if (t < N * IN_CH) s1[t] = x[t];
    if (t < N)         deg[t] = 1.0f;
}

// ---------------------------------------------------------------------------
// Kernel 2: in-degree accumulation (float atomics; counts < 2^24 are exact).
// edge_index is int64, shape [2,E] row-major: src at [e], dst at [E+e].
// ---------------------------------------------------------------------------
__global__ void k_deg(const long long* __restrict__ ei, float* __restrict__ deg,
                      int E) {
    int e = blockIdx.x * blockDim.x + threadIdx.x;
    if (e < E) atomicAdd(deg + (int)ei[E + e], 1.0f);
}

// ---------------------------------------------------------------------------
// Kernel 3: scatter layer-1 features: s1[dst,c] += x[src,c].  E*32 threads.
// ---------------------------------------------------------------------------
__global__ void k_scatter1(const float* __restrict__ x,
                           const long long* __restrict__ ei,
                           float* __restrict__ s1, int E) {
    int t = blockIdx.x * blockDim.x + threadIdx.x;
    if (t >= E * IN_CH) return;
    int e = t >> 5;          // IN_CH == 32
    int c = t & 31;
    int s = (int)ei[e];
    int d = (int)ei[E + e];
    atomicAdd(&s1[d * IN_CH + c], x[s * IN_CH + c]);
}

// ---------------------------------------------------------------------------
// Kernel 4: deg -> 1/deg  (deg >= 1 always, self loop guarantees no div0)
// ---------------------------------------------------------------------------
__global__ void k_recip(float* __restrict__ deg, int N) {
    int t = blockIdx.x * blockDim.x + threadIdx.x;
    if (t < N) deg[t] = 1.0f / deg[t];
}

// ---------------------------------------------------------------------------
// Kernel 5: WMMA fp32 GEMM:  h1 = relu((s1 * rdeg_row) @ W1 + b1)
//   A: [16 nodes x 32]  B: W1 [32 x 400]  -> one wave per 16x16 output tile,
//   K=32 as 8x V_WMMA_F32_16X16X4_F32.
// VGPR layouts per CDNA5 ISA 7.12.2:
//   A 16x4 : lane l=lane&15, h=lane>>4; frag[v] = A[M=l][K = kbase + 2h + v]
//   B 4x16 : frag[v] = B[K = kbase + 2h + v][N=l]
//   C/D    : acc[v]  = D[M = v + 8h][N = l]
// WMMA requires EXEC all-ones: no divergent control flow around the WMMA;
// out-of-range rows/cols are clamp-loaded and store-guarded afterwards.
// ---------------------------------------------------------------------------
__global__ void k_gemm1(const float* __restrict__ s1,
                        const float* __restrict__ rdeg,
                        const float* __restrict__ W1,
                        const float* __restrict__ b1,
                        float* __restrict__ h1, int N) {
    const int lane = threadIdx.x;        // blockDim.x == 32 (one wave)
    const int l    = lane & 15;
    const int h    = lane >> 4;          // 0 or 1
    const int n0   = blockIdx.x * 16;    // node tile
    const int c0   = blockIdx.y * 16;    // output-channel tile

    int arow_n = n0 + l;  if (arow_n >= N)   arow_n = N - 1;      // clamp
    int bcol_n = c0 + l;  if (bcol_n >= HID) bcol_n = HID - 1;    // clamp

    const float scale = rdeg[arow_n];                 // 1/deg for this A row
    const float* __restrict__ arow = s1 + (size_t)arow_n * IN_CH + 2 * h;
    const float* __restrict__ bcol = W1 + bcol_n;

    v8f acc = {};
#pragma unroll
    for (int kk = 0; kk < IN_CH / 4; ++kk) {
        const int kb = kk * 4;
        v2f a, b;
        a.x = arow[kb + 0] * scale;                   // A[l][kb+2h]
        a.y = arow[kb + 1] * scale;                   // A[l][kb+2h+1]
        b.x = bcol[(size_t)(kb + 2 * h + 0) * HID];   // B[kb+2h  ][l]
        b.y = bcol[(size_t)(kb + 2 * h + 1) * HID];   // B[kb+2h+1][l]
        acc = __builtin_amdgcn_wmma_f32_16x16x4_f32(
            /*neg_a=*/false, a, /*neg_b=*/false, b,
            /*c_mod=*/(short)0, acc, /*reuse_a=*/false, /*reuse_b=*/false);
    }

    const float bias = b1[bcol_n];
#pragma unroll
    for (int v = 0; v < 8; ++v) {
        const int m = n0 + v + 8 * h;                 // node row for acc[v]
        const int c = c0 + l;
        if (m < N && c < HID) {
            float val = acc[v] + bias;
            h1[(size_t)m * HID + c] = val > 0.0f ? val : 0.0f;
        }
    }
}

// ---------------------------------------------------------------------------
// Kernel 6: layer-2 self loop: s2 = h1 (copy).  N*400 threads.
// ---------------------------------------------------------------------------
__global__ void k_init2(const float* __restrict__ h1, float* __restrict__ s2,
                        int total) {
    int t = blockIdx.x * blockDim.x + threadIdx.x;
    if (t < total) s2[t] = h1[t];
}

// ---------------------------------------------------------------------------
// Kernel 7: scatter layer-2 features: s2[dst,c] += h1[src,c].  E*400 threads.
// h1 (80MB) and s2 (80MB) both fit in the 192MB L2 -> atomics stay on-die.
// ---------------------------------------------------------------------------
__global__ void k_scatter2(const float* __restrict__ h1,
                           const long long* __restrict__ ei,
                           float* __restrict__ s2, int E) {
    int t = blockIdx.x * blockDim.x + threadIdx.x;
    if (t >= E * HID) return;
    int e = t / HID;
    int c = t - e * HID;
    int s = (int)ei[e];
    int d = (int)ei[E + e];
    atomicAdd(&s2[d * HID + c], h1[s * HID + c]);
}

// ---------------------------------------------------------------------------
// Kernel 8: out = relu((s2 * rdeg_row) @ W2 + b2).  400 -> 4, one thread per
// node; W2 (1600 floats) staged in LDS; float4 row loads.
// ---------------------------------------------------------------------------
__global__ void k_gemm2(const float* __restrict__ s2,
                        const float* __restrict__ rdeg,
                        const float* __restrict__ W2,
                        const float* __restrict__ b2,
                        float* __restrict__ out, int N) {
    __shared__ float w[HID * OUT_CH];
    for (int i = threadIdx.x; i < HID * OUT_CH; i += blockDim.x) w[i] = W2[i];
    __syncthreads();

    int n = blockIdx.x * blockDim.x + threadIdx.x;
    if (n >= N) return;

    const float r = rdeg[n];
    const float4* __restrict__ row = (const float4*)(s2 + (size_t)n * HID);
    float a0 = 0.f, a1 = 0.f, a2 = 0.f, a3 = 0.f;
#pragma unroll 4
    for (int k4 = 0; k4 < HID / 4; ++k4) {
        float4 v = row[k4];
        int kb = k4 * 4;
        a0 += v.x * w[(kb + 0) * 4 + 0] + v.y * w[(kb + 1) * 4 + 0] +
              v.z * w[(kb + 2) * 4 + 0] + v.w * w[(kb + 3) * 4 + 0];
        a1 += v.x * w[(kb + 0) * 4 + 1] + v.y * w[(kb + 1) * 4 + 1] +
              v.z * w[(kb + 2) * 4 + 1] + v.w * w[(kb + 3) * 4 + 1];
        a2 += v.x * w[(kb + 0) * 4 + 2] + v.y * w[(kb + 1) * 4 + 2] +
              v.z * w[(kb + 2) * 4 + 2] + v.w * w[(kb + 3) * 4 + 2];
        a3 += v.x * w[(kb + 0) * 4 + 3] + v.y * w[(kb + 1) * 4 + 3] +
              v.z * w[(kb + 2) * 4 + 3] + v.w * w[(kb + 3) * 4 + 3];
    }
    a0 = a0 * r + b2[0];
    a1 = a1 * r + b2[1];
    a2 = a2 * r + b2[2];
    a3 = a3 * r + b2[3];
    float* o = out + (size_t)n * OUT_CH;
    o[0] = a0 > 0.f ? a0 : 0.f;
    o[1] = a1 > 0.f ? a1 : 0.f;
    o[2] = a2 > 0.f ? a2 : 0.f;
    o[3] = a3 > 0.f ? a3 : 0.f;
}

// ---------------------------------------------------------------------------
// Host-side launcher.  HEADS==1 makes softmax identically 1, so u1/c1/u2/c2
// (d_in[3],[4],[7],[8]) are mathematically dead.
// ---------------------------------------------------------------------------
extern "C" void kernel_launch(void* const* d_in, const int* in_sizes, int n_in,
                              void* d_out, int out_size, void* d_ws, size_t ws_size,
                              hipStream_t stream) {
    const float*     x  = (const float*)d_in[0];
    const long long* ei = (const long long*)d_in[1];   // int64 edge_index [2,E]
    const float*     W1 = (const float*)d_in[2];
    const float*     b1 = (const float*)d_in[5];
    const float*     W2 = (const float*)d_in[6];
    const float*     b2 = (const float*)d_in[9];
    float*           out = (float*)d_out;

    const int N = in_sizes[0] / IN_CH;
    const int E = in_sizes[1] / 2;

    // Workspace layout (floats): deg[N] | s1[N*32] | h1[N*400] | s2[N*400]
    float* deg = (float*)d_ws;
    float* s1  = deg + N;
    float* h1  = s1 + (size_t)N * IN_CH;
    float* s2  = h1 + (size_t)N * HID;

    const int B = 256;

    k_init1<<<(N * IN_CH + B - 1) / B, B, 0, stream>>>(x, s1, deg, N);
    k_deg<<<(E + B - 1) / B, B, 0, stream>>>(ei, deg, E);
    k_scatter1<<<(E * IN_CH + B - 1) / B, B, 0, stream>>>(x, ei, s1, E);
    k_recip<<<(N + B - 1) / B, B, 0, stream>>>(deg, N);

    dim3 g1((N + 15) / 16, (HID + 15) / 16);
    k_gemm1<<<g1, 32, 0, stream>>>(s1, deg, W1, b1, h1, N);

    k_init2<<<(N * HID + B - 1) / B, B, 0, stream>>>(h1, s2, N * HID);
    k_scatter2<<<(E * HID + B - 1) / B, B, 0, stream>>>(h1, ei, s2, E);
    k_gemm2<<<(N + B - 1) / B, B, 0, stream>>>(s2, deg, W2, b2, out, N);
}